// TopKRouter_33852932227538
// MI455X (gfx1250) — compile-verified
//
#include <hip/hip_runtime.h>
#include <hip/hip_bf16.h>
#include <math.h>

typedef __attribute__((ext_vector_type(2))) float v2f;
typedef __attribute__((ext_vector_type(8))) float v8f;

#define D_MODEL     2048
#define NUM_EXPERTS 64
#define ROWS_PER_WG 128
#define KC          64
#define NCHUNK      (D_MODEL / KC)   // 32
#define XSTRIDE     68   // dwords; 68 % 64 == 4 -> conflict-free 16x4 fragment reads
#define WSTRIDE     68
#define LSTRIDE     65   // logits staging stride

// ---- CDNA5 async global->LDS copy (ASYNCcnt-tracked, no VGPR round trip) ----
// VDST = per-lane LDS byte offset (low 32 bits of generic shared pointer ==
// allocation-relative LDS offset per the flat-addressing aperture rules),
// VADDR = 64-bit global address.
__device__ __forceinline__ void async_copy_b128(const float* g, float* l) {
    unsigned loff = (unsigned)(unsigned long long)l;
    asm volatile("global_load_async_to_lds_b128 %0, %1, off"
                 :: "v"(loff), "v"(g) : "memory");
}

__device__ __forceinline__ void issue_chunk(const float* __restrict__ x,
                                            const float* __restrict__ Wg,
                                            int row0, int kc, int t,
                                            float* xbuf, float* wbuf) {
    // x tile: 128 rows x 64 f32, 16 float4/row, fully coalesced (8 ops/thread)
    #pragma unroll
    for (int it = 0; it < 8; ++it) {
        int i  = t + 256 * it;
        int r  = i >> 4;
        int c4 = i & 15;
        async_copy_b128(x + (size_t)(row0 + r) * D_MODEL + kc + c4 * 4,
                        &xbuf[r * XSTRIDE + c4 * 4]);
    }
    // W tile: 64 experts x 64 f32 (4 ops/thread), L2-resident after first touch
    #pragma unroll
    for (int it = 0; it < 4; ++it) {
        int i  = t + 256 * it;
        int e  = i >> 4;
        int c4 = i & 15;
        async_copy_b128(Wg + (size_t)e * D_MODEL + kc + c4 * 4,
                        &wbuf[e * WSTRIDE + c4 * 4]);
    }
}

__global__ __launch_bounds__(256) void
router_topk_kernel(const float* __restrict__ x,
                   const float* __restrict__ Wg,
                   float* __restrict__ out, int N)
{
    __shared__ float xs[2][ROWS_PER_WG * XSTRIDE];   // 2 x 34816 B (double buffer)
    __shared__ float ws[2][NUM_EXPERTS * WSTRIDE];   // 2 x 17408 B
    float* ls = &xs[0][0];                            // logits reuse: 128*65 <= 128*68

    const int t     = threadIdx.x;
    const int lane  = t & 31;
    const int wave  = t >> 5;
    const int row0  = blockIdx.x * ROWS_PER_WG;

    const int m     = lane & 15;   // row/column within fragment
    const int khalf = lane >> 4;   // which K pair this half-wave holds

    v8f acc0 = {}, acc1 = {}, acc2 = {}, acc3 = {};

    // prime the pipeline: chunk 0 -> buffer 0 (12 async ops per thread)
    issue_chunk(x, Wg, row0, 0, t, &xs[0][0], &ws[0][0]);

    for (int ci = 0; ci < NCHUNK; ++ci) {
        const int cur = ci & 1;

        // prefetch chunk ci+1 into the other buffer (last read in iter ci-1,
        // protected by that iteration's trailing barrier)
        if (ci + 1 < NCHUNK) {
            issue_chunk(x, Wg, row0, (ci + 1) * KC, t, &xs[cur ^ 1][0], &ws[cur ^ 1][0]);
            // async loads complete in order: <=12 outstanding => chunk ci arrived
            asm volatile("s_wait_asynccnt 0xc" ::: "memory");
        } else {
            asm volatile("s_wait_asynccnt 0x0" ::: "memory");
        }
        __syncthreads();   // all waves' portions of chunk ci visible in LDS

        const float* xw = &xs[cur][(wave * 16 + m) * XSTRIDE];
        const float* wb = &ws[cur][0];
        #pragma unroll
        for (int kk = 0; kk < KC; kk += 4) {
            // A: 16x4 f32 fragment of this wave's 16 rows
            v2f a = *(const v2f*)(xw + kk + 2 * khalf);
            // B: 4x16 f32 fragments of W^T, one per 16-expert group
            v2f b0 = *(const v2f*)(wb + ( 0 + m) * WSTRIDE + kk + 2 * khalf);
            v2f b1 = *(const v2f*)(wb + (16 + m) * WSTRIDE + kk + 2 * khalf);
            v2f b2 = *(const v2f*)(wb + (32 + m) * WSTRIDE + kk + 2 * khalf);
            v2f b3 = *(const v2f*)(wb + (48 + m) * WSTRIDE + kk + 2 * khalf);
            acc0 = __builtin_amdgcn_wmma_f32_16x16x4_f32(false, a, false, b0, (short)0, acc0, false, false);
            acc1 = __builtin_amdgcn_wmma_f32_16x16x4_f32(false, a, false, b1, (short)0, acc1, false, false);
            acc2 = __builtin_amdgcn_wmma_f32_16x16x4_f32(false, a, false, b2, (short)0, acc2, false, false);
            acc3 = __builtin_amdgcn_wmma_f32_16x16x4_f32(false, a, false, b3, (short)0, acc3, false, false);
        }
        __syncthreads();   // done reading buf cur before iter ci+1 overwrites it
    }

    // ---- scatter accumulators to LDS: row-major 128 x 64 logits (stride 65) ----
    {
        int rbase = wave * 16 + 8 * khalf;  // VGPR j holds row j + 8*khalf, col m
        #pragma unroll
        for (int j = 0; j < 8; ++j) {
            float* lr = &ls[(rbase + j) * LSTRIDE];
            lr[ 0 + m] = acc0[j];
            lr[16 + m] = acc1[j];
            lr[32 + m] = acc2[j];
            lr[48 + m] = acc3[j];
        }
    }
    __syncthreads();

    float* out_w = out;                     // (N,2) top-k weights
    float* out_i = out + (size_t)2 * N;     // (N,2) indices (as float)
    float* out_l = out + (size_t)4 * N;     // (N,64) logits

    // ---- one thread per row: softmax + top-2 + renormalize ----
    if (t < ROWS_PER_WG) {
        const float* lr = &ls[t * LSTRIDE];
        float mx = -INFINITY;
        #pragma unroll 8
        for (int e = 0; e < NUM_EXPERTS; ++e) mx = fmaxf(mx, lr[e]);
        float s = 0.0f;
        #pragma unroll 8
        for (int e = 0; e < NUM_EXPERTS; ++e) s += __expf(lr[e] - mx);

        float b1 = -INFINITY, b2 = -INFINITY;
        int   i1 = 0,         i2 = 0;
        for (int e = 0; e < NUM_EXPERTS; ++e) {
            float v = lr[e];
            if (v > b1)      { b2 = b1; i2 = i1; b1 = v; i1 = e; }
            else if (v > b2) { b2 = v; i2 = e; }
        }
        float p1 = __expf(b1 - mx) / s;
        float p2 = __expf(b2 - mx) / s;
        float inv = 1.0f / (p1 + p2 + 1e-10f);

        size_t grow = (size_t)(row0 + t);
        out_w[grow * 2 + 0] = p1 * inv;
        out_w[grow * 2 + 1] = p2 * inv;
        out_i[grow * 2 + 0] = (float)i1;
        out_i[grow * 2 + 1] = (float)i2;
    }

    // ---- coalesced logits writeback: 128 x 64 floats ----
    #pragma unroll
    for (int it = 0; it < 32; ++it) {
        int i = t + 256 * it;              // 0..8191
        int r = i >> 6;
        int e = i & 63;
        out_l[((size_t)(row0 + r)) * NUM_EXPERTS + e] = ls[r * LSTRIDE + e];
    }
}

extern "C" void kernel_launch(void* const* d_in, const int* in_sizes, int n_in,
                              void* d_out, int out_size, void* d_ws, size_t ws_size,
                              hipStream_t stream) {
    const float* x  = (const float*)d_in[0];
    const float* Wg = (const float*)d_in[1];
    float* out = (float*)d_out;
    int N = in_sizes[0] / D_MODEL;          // 16384 rows
    int blocks = N / ROWS_PER_WG;           // 128 workgroups
    router_topk_kernel<<<blocks, 256, 0, stream>>>(x, Wg, out, N);
}